// GLM_Encoder_81140522156036
// MI455X (gfx1250) — compile-verified
//
#include <hip/hip_runtime.h>
#include <hip/hip_bf16.h>
#include <math.h>

typedef float v2f __attribute__((ext_vector_type(2)));
typedef float v8f __attribute__((ext_vector_type(8)));

#define T_DATA 50000
#define E_NO   2000
#define I_NO   500
#define NCH    64     // 63 channels padded to 64
#define GTM    32     // GEMM M tile per block
#define GKB    64     // GEMM K tile per stage
#define GSA    68     // A LDS row stride (floats) - conflict free for b64 A reads
#define GSB    80     // B LDS row stride (floats) - conflict free for split-half B reads
#define CT     64     // conv t tile

// ---------------------------------------------------------------------------
// Build Bt[k*64 + c] = C_syn[(c+1)*K + k], padded col 63 = 0
// ---------------------------------------------------------------------------
__global__ void prep_bt_kernel(const float* __restrict__ Csyn, int K,
                               float* __restrict__ Bt) {
  int idx = blockIdx.x * blockDim.x + threadIdx.x;
  if (idx >= K * NCH) return;
  int k = idx >> 6;
  int c = idx & 63;
  Bt[idx] = (c < 63) ? Csyn[(size_t)(c + 1) * K + k] : 0.f;
}

// ---------------------------------------------------------------------------
// Evaluate raised-cosine bases (fp64, matching numpy) and project:
//   KeT[k*64+c] = sum_b W_syn[c,b,0]*cos_basis[b,k]   (k=0..199)
//   KiT[k*64+c] = sum_b W_syn[c,b,1]*cos_basis[b,k]
//   OKT[s*64+c] = sum_d W_obs[c,d]*obs_basis[d,s]     (s=0..400, unflipped)
// grid: 401 blocks x 64 threads (blockIdx = tap index, thread = channel)
// ---------------------------------------------------------------------------
__global__ void prep_filters_kernel(const float* __restrict__ W_syn,
                                    const float* __restrict__ W_obs,
                                    float* __restrict__ KeT,
                                    float* __restrict__ KiT,
                                    float* __restrict__ OKT) {
  const int s = blockIdx.x;   // 0..400
  const int c = threadIdx.x;  // 0..63
  const double PI = 3.14159265358979323846;

  if (s < 200) {
    float ke = 0.f, ki = 0.f;
    if (c < 63) {
      double raw = 5.0 * log((double)s + 1.0 + 1e-8);
      for (int b = 0; b < 13; ++b) {
        double phi = 0.5 * PI * (double)b;
        double v = 0.0;
        if (!(raw < phi - PI || raw > phi + PI)) v = 0.5 * cos(raw - phi) + 0.5;
        float vf = (float)v;
        ke += W_syn[(c * 13 + b) * 2 + 0] * vf;
        ki += W_syn[(c * 13 + b) * 2 + 1] * vf;
      }
    }
    KeT[s * NCH + c] = ke;
    KiT[s * NCH + c] = ki;
  }

  float ok = 0.f;
  if (c < 63) {
    int x = s - 200;
    double rawo = 5.0 * log(fabs((double)x) + 1.0 + 1e-8);
    for (int d = 0; d < 25; ++d) {
      int i;
      int side;
      if (d == 0)       { i = 0;           side = 1; }
      else if (d & 1)   { i = (d + 1) / 2; side = (s >= 200); }
      else              { i = d / 2;       side = (s <= 200); }
      double v = 0.0;
      if (side) {
        double phi = 0.5 * PI * (double)i;
        if (!(rawo < phi - PI || rawo > phi + PI)) v = 0.5 * cos(rawo - phi) + 0.5;
      }
      ok += W_obs[c * 25 + d] * (float)v;
    }
  }
  OKT[s * NCH + c] = ok;
}

// ---------------------------------------------------------------------------
// GEMM: Out[m, 0..63] = A[m, :] @ Bt[:, 0..63] using V_WMMA_F32_16X16X4_F32.
// Block: 256 threads = 8 waves -> 2 M-subtiles x 4 N-subtiles of 16x16.
// ---------------------------------------------------------------------------
__global__ __launch_bounds__(256) void gemm_wmma_kernel(
    const float* __restrict__ A, const float* __restrict__ Bt,
    int M, int K, float* __restrict__ Out) {
  __shared__ float lA[GTM * GSA];   // 8.7 KB
  __shared__ float lB[GKB * GSB];   // 20.5 KB

  const int tid  = threadIdx.x;
  const int lane = tid & 31;
  const int wave = tid >> 5;
  const int msub = wave & 1;
  const int nsub = wave >> 1;
  const int m0   = blockIdx.x * GTM;
  const int half = lane >> 4;       // which K-pair half this lane holds
  const int lr   = lane & 15;

  v8f acc = {0.f, 0.f, 0.f, 0.f, 0.f, 0.f, 0.f, 0.f};

  const float* aptr  = &lA[(msub * 16 + lr) * GSA + 2 * half];
  const int    nbase = nsub * 16 + lr;

  for (int k0 = 0; k0 < K; k0 += GKB) {
    const int kb = (K - k0 < GKB) ? (K - k0) : GKB;
    __syncthreads();
    // stage A tile (GTM x GKB) -- 2 float4 per thread
#pragma unroll
    for (int i = 0; i < 2; ++i) {
      int slot = tid + i * 256;          // 0..511
      int row  = slot >> 4;
      int kk   = (slot & 15) * 4;
      float4 v = make_float4(0.f, 0.f, 0.f, 0.f);
      int gr = m0 + row;
      if (gr < M && kk < kb)             // kb is always a multiple of 4 here
        v = *(const float4*)(A + (size_t)gr * K + k0 + kk);
      *(float4*)&lA[row * GSA + kk] = v;
    }
    // stage B tile (GKB x 64) -- 4 float4 per thread
#pragma unroll
    for (int i = 0; i < 4; ++i) {
      int slot = tid + i * 256;          // 0..1023
      int row  = slot >> 4;
      int n    = (slot & 15) * 4;
      float4 v = make_float4(0.f, 0.f, 0.f, 0.f);
      if (row < kb)
        v = *(const float4*)(Bt + (size_t)(k0 + row) * NCH + n);
      *(float4*)&lB[row * GSB + n] = v;
    }
    __syncthreads();

    if (kb == GKB) {
#pragma unroll
      for (int kk = 0; kk < GKB; kk += 4) {
        v2f a = *(const v2f*)(aptr + kk);
        v2f b;
        b.x = lB[(kk + 2 * half) * GSB + nbase];
        b.y = lB[(kk + 1 + 2 * half) * GSB + nbase];
        acc = __builtin_amdgcn_wmma_f32_16x16x4_f32(
            false, a, false, b, (short)0, acc, false, false);
      }
    } else {
      for (int kk = 0; kk < kb; kk += 4) {
        v2f a = *(const v2f*)(aptr + kk);
        v2f b;
        b.x = lB[(kk + 2 * half) * GSB + nbase];
        b.y = lB[(kk + 1 + 2 * half) * GSB + nbase];
        acc = __builtin_amdgcn_wmma_f32_16x16x4_f32(
            false, a, false, b, (short)0, acc, false, false);
      }
    }
  }

  // D layout: VGPR r -> M = r (lanes 0-15), r+8 (lanes 16-31); N = lr
  const int row0 = m0 + msub * 16;
#pragma unroll
  for (int r = 0; r < 8; ++r) {
    int grow = row0 + r + 8 * half;
    if (grow < M) Out[(size_t)grow * NCH + nbase] = acc[r];
  }
}

// ---------------------------------------------------------------------------
// Depthwise causal convs (200-tap e + i) + 401-tap obs correlation + fused
// Gumbel softmax / sigmoid outputs. Block: 256 threads = 64 ch x 4 t-groups,
// each thread accumulates 16 timesteps (taps amortized 16x).
// ---------------------------------------------------------------------------
__global__ __launch_bounds__(256) void conv_gumbel_kernel(
    const float* __restrict__ syn_e, const float* __restrict__ syn_i,
    const float* __restrict__ Z, const float* __restrict__ KeT,
    const float* __restrict__ KiT, const float* __restrict__ OKT,
    const float* __restrict__ Theta, const float* __restrict__ temp_p,
    const float* __restrict__ u, float* __restrict__ out, int T) {
  __shared__ float se[(CT + 199) * NCH];   // 67.3 KB
  __shared__ float si[(CT + 199) * NCH];   // 67.3 KB
  __shared__ float zw[CT + 400];

  const int tid = threadIdx.x;
  const int t0  = blockIdx.x * CT;

  // stage syn windows (rows t0-199 .. t0+CT-1) and Z window
  for (int slot = tid; slot < (CT + 199) * 16; slot += 256) {
    int row = slot >> 4;
    int c4  = (slot & 15) * 4;
    int t   = t0 - 199 + row;
    float4 ve = make_float4(0.f, 0.f, 0.f, 0.f);
    float4 vi = make_float4(0.f, 0.f, 0.f, 0.f);
    if (t >= 0 && t < T) {
      ve = *(const float4*)(syn_e + (size_t)t * NCH + c4);
      vi = *(const float4*)(syn_i + (size_t)t * NCH + c4);
    }
    *(float4*)&se[row * NCH + c4] = ve;
    *(float4*)&si[row * NCH + c4] = vi;
  }
  for (int iz = tid; iz < CT + 400; iz += 256) {
    int t = t0 - 200 + iz;
    zw[iz] = (t >= 0 && t < T) ? Z[t] : 0.f;
  }
  __syncthreads();

  const int c  = tid & 63;
  const int gq = tid >> 6;       // 0..3
  const int tb = gq * 16;        // local t base

  float acc[16];
#pragma unroll
  for (int j = 0; j < 16; ++j) acc[j] = 0.f;

  // filt[t,c] = sum_k syn[t-k,c] * K[c,k]
  for (int k = 0; k < 200; ++k) {
    float we = KeT[k * NCH + c];
    float wi = KiT[k * NCH + c];
    int base = (tb + 199 - k) * NCH + c;
#pragma unroll
    for (int j = 0; j < 16; ++j) {
      acc[j] += we * se[base + j * NCH];
      acc[j] += wi * si[base + j * NCH];
    }
  }
  // Z_filt[t,c] = sum_s Z[t+200-s] * OK[c,s]
  for (int s = 0; s <= 400; ++s) {
    float wo = OKT[s * NCH + c];
    int base = tb + 400 - s;
#pragma unroll
    for (int j = 0; j < 16; ++j) acc[j] += wo * zw[base + j];
  }

  const float temp    = *temp_p;
  const float invtemp = 1.f / temp;
  const float th      = (c < 63) ? Theta[c] : 0.f;

  if (c < 63) {
#pragma unroll
    for (int j = 0; j < 16; ++j) {
      int t = t0 + tb + j;
      if (t < T) {
        float L0 = acc[j] + th;
        size_t bi = (size_t)t * 63 + c;
        float u0 = u[bi * 2 + 0];
        float u1 = u[bi * 2 + 1];
        float g0 = -logf(-logf(u0 + 1e-8f) + 1e-8f);
        float g1 = -logf(-logf(u1 + 1e-8f) + 1e-8f);
        float zh = 1.f / (1.f + expf(-(L0 + g0 - g1) * invtemp));
        float sg = 1.f / (1.f + expf(-L0));
        out[bi] = zh;
        out[(size_t)T * 63 + bi] = sg;
      }
    }
  }
}

// ---------------------------------------------------------------------------
extern "C" void kernel_launch(void* const* d_in, const int* in_sizes, int n_in,
                              void* d_out, int out_size, void* d_ws,
                              size_t ws_size, hipStream_t stream) {
  const float* S_e   = (const float*)d_in[0];
  const float* S_i   = (const float*)d_in[1];
  const float* Z_obs = (const float*)d_in[2];
  const float* temp  = (const float*)d_in[3];
  const float* u     = (const float*)d_in[4];
  const float* C_e   = (const float*)d_in[5];
  const float* C_i   = (const float*)d_in[6];
  const float* W_syn = (const float*)d_in[7];
  const float* W_obs = (const float*)d_in[8];
  const float* Theta = (const float*)d_in[9];
  float* out = (float*)d_out;

  const int T = T_DATA;
  float* ws = (float*)d_ws;
  size_t off = 0;
  float* syn_e = ws + off;  off += (size_t)T * NCH;      // 3.2M
  float* syn_i = ws + off;  off += (size_t)T * NCH;      // 3.2M
  float* Bt_e  = ws + off;  off += (size_t)E_NO * NCH;   // 128K
  float* Bt_i  = ws + off;  off += (size_t)I_NO * NCH;   // 32K
  float* KeT   = ws + off;  off += 200 * NCH;            // 12.8K
  float* KiT   = ws + off;  off += 200 * NCH;            // 12.8K
  float* OKT   = ws + off;  off += 401 * NCH;            // 25.7K

  prep_bt_kernel<<<(E_NO * NCH + 255) / 256, 256, 0, stream>>>(C_e, E_NO, Bt_e);
  prep_bt_kernel<<<(I_NO * NCH + 255) / 256, 256, 0, stream>>>(C_i, I_NO, Bt_i);
  prep_filters_kernel<<<401, 64, 0, stream>>>(W_syn, W_obs, KeT, KiT, OKT);

  gemm_wmma_kernel<<<(T + GTM - 1) / GTM, 256, 0, stream>>>(S_e, Bt_e, T, E_NO, syn_e);
  gemm_wmma_kernel<<<(T + GTM - 1) / GTM, 256, 0, stream>>>(S_i, Bt_i, T, I_NO, syn_i);

  conv_gumbel_kernel<<<(T + CT - 1) / CT, 256, 0, stream>>>(
      syn_e, syn_i, Z_obs, KeT, KiT, OKT, Theta, temp, u, out, T);
}